// AdaptiveTokenFilter_62697932587509
// MI455X (gfx1250) — compile-verified
//
#include <hip/hip_runtime.h>
#include <math.h>

typedef __attribute__((ext_vector_type(2))) float v2f;
typedef __attribute__((ext_vector_type(8))) float v8f;

#define BB 8
#define SS 8192
#define EE 1024
#define HH 64
#define LDS_STRIDE 1028  // 64 rows x 1028 floats (padded) = 263168 B < 320 KB

// ---------------------------------------------------------------------------
// Kernel 1: logits = relu(TE @ W1 + b1) @ W2 + b2   via V_WMMA_F32_16X16X4_F32
// grid 512 x 256 threads (8 waves). Each wave: 16 tokens x 64 hidden.
// ---------------------------------------------------------------------------
__global__ __launch_bounds__(256) void k_logits(
    const float* __restrict__ te, const float* __restrict__ w1,
    const float* __restrict__ bias1, const float* __restrict__ w2,
    const float* __restrict__ bias2, float* __restrict__ logits)
{
    extern __shared__ float ldsT[];  // W1 transposed: [H][LDS_STRIDE]
    const int tid = threadIdx.x;

    // cooperative load + transpose of W1 (E x H row-major -> H x E padded)
    for (int i = tid; i < (EE * HH / 4); i += 256) {
        const float4 v = ((const float4*)w1)[i];
        const int k  = i >> 4;          // E index (16 float4 per W1 row)
        const int n0 = (i & 15) * 4;    // H index
        ldsT[(n0 + 0) * LDS_STRIDE + k] = v.x;
        ldsT[(n0 + 1) * LDS_STRIDE + k] = v.y;
        ldsT[(n0 + 2) * LDS_STRIDE + k] = v.z;
        ldsT[(n0 + 3) * LDS_STRIDE + k] = v.w;
    }
    __syncthreads();

    const int lane  = tid & 31;
    const int wave  = tid >> 5;
    const int m     = lane & 15;          // token-within-tile / N-within-tile
    const int khalf = (lane >> 4) << 1;   // 0 or 2 (K pair select per ISA layout)
    const int tg0   = blockIdx.x * 128 + wave * 16;  // first flat token of tile

    const float* arow = te + (size_t)(tg0 + m) * EE + khalf;
    const float* f0p = ldsT + ( 0 + m) * LDS_STRIDE + khalf;
    const float* f1p = ldsT + (16 + m) * LDS_STRIDE + khalf;
    const float* f2p = ldsT + (32 + m) * LDS_STRIDE + khalf;
    const float* f3p = ldsT + (48 + m) * LDS_STRIDE + khalf;

    v8f acc0 = {}, acc1 = {}, acc2 = {}, acc3 = {};
    #pragma unroll 4
    for (int k = 0; k < EE; k += 4) {
        const v2f a  = *(const v2f*)(arow + k);
        const v2f f0 = *(const v2f*)(f0p + k);
        const v2f f1 = *(const v2f*)(f1p + k);
        const v2f f2 = *(const v2f*)(f2p + k);
        const v2f f3 = *(const v2f*)(f3p + k);
        acc0 = __builtin_amdgcn_wmma_f32_16x16x4_f32(false, a, false, f0, (short)0, acc0, false, false);
        acc1 = __builtin_amdgcn_wmma_f32_16x16x4_f32(false, a, false, f1, (short)0, acc1, false, false);
        acc2 = __builtin_amdgcn_wmma_f32_16x16x4_f32(false, a, false, f2, (short)0, acc2, false, false);
        acc3 = __builtin_amdgcn_wmma_f32_16x16x4_f32(false, a, false, f3, (short)0, acc3, false, false);
    }

    // epilogue: bias + relu + dot with W2, fused in registers.
    // C layout: VGPR r, lane -> M = r + 8*(lane/16), N = t*16 + (lane%16)
    float part[8];
    #pragma unroll
    for (int r = 0; r < 8; ++r) part[r] = 0.0f;

#define EPILOG(ACC, T)                                          \
    {                                                           \
        const int   n  = (T) * 16 + m;                          \
        const float bb = bias1[n];                              \
        const float ww = w2[n];                                 \
        _Pragma("unroll")                                       \
        for (int r = 0; r < 8; ++r) {                           \
            float h = ACC[r] + bb;                              \
            h = h > 0.0f ? h : 0.0f;                            \
            part[r] += h * ww;                                  \
        }                                                       \
    }
    EPILOG(acc0, 0) EPILOG(acc1, 1) EPILOG(acc2, 2) EPILOG(acc3, 3)
#undef EPILOG

    // reduce across the 16 lanes of each half-wave (N dimension)
    #pragma unroll
    for (int r = 0; r < 8; ++r) {
        float v = part[r];
        v += __shfl_xor(v, 1, 32);
        v += __shfl_xor(v, 2, 32);
        v += __shfl_xor(v, 4, 32);
        v += __shfl_xor(v, 8, 32);
        part[r] = v;
    }
    if (m == 0) {  // lanes 0 and 16 hold tokens [0..7] and [8..15] of the tile
        const float b2v = bias2[0];
        const int mb = (lane >> 4) * 8;
        #pragma unroll
        for (int r = 0; r < 8; ++r)
            logits[tg0 + mb + r] = part[r] + b2v;
    }
}

// ---------------------------------------------------------------------------
// Kernel 2a: per batch row -> expected_k, k, z = logit + gumbel, max(z), sum(exp)
// ---------------------------------------------------------------------------
__global__ __launch_bounds__(1024) void k_row_stats(
    const float* __restrict__ logits, const float* __restrict__ u,
    float* __restrict__ z, float* __restrict__ rowmax, float* __restrict__ rowsum,
    int* __restrict__ kcnt, float* __restrict__ expected_out)
{
    __shared__ float sd[1024];
    const int b = blockIdx.x;
    const int tid = threadIdx.x;
    const float* lg = logits + (size_t)b * SS;
    const float* uu = u + (size_t)b * SS;
    float* zz = z + (size_t)b * SS;

    float sig = 0.0f;
    float mx  = -INFINITY;
    for (int s = tid; s < SS; s += 1024) {
        const float l  = lg[s];
        const float g  = -logf(-logf(uu[s]));
        const float zv = l + g;  // TAU == 1.0
        zz[s] = zv;
        sig += 1.0f / (1.0f + expf(-l));
        mx = fmaxf(mx, zv);
    }
    sd[tid] = sig; __syncthreads();
    for (int o = 512; o > 0; o >>= 1) { if (tid < o) sd[tid] += sd[tid + o]; __syncthreads(); }
    const float sigsum = sd[0]; __syncthreads();

    sd[tid] = mx; __syncthreads();
    for (int o = 512; o > 0; o >>= 1) { if (tid < o) sd[tid] = fmaxf(sd[tid], sd[tid + o]); __syncthreads(); }
    const float zmax = sd[0]; __syncthreads();

    float se = 0.0f;
    for (int s = tid; s < SS; s += 1024) se += expf(zz[s] - zmax);
    sd[tid] = se; __syncthreads();
    for (int o = 512; o > 0; o >>= 1) { if (tid < o) sd[tid] += sd[tid + o]; __syncthreads(); }

    if (tid == 0) {
        rowmax[b] = zmax;
        rowsum[b] = sd[0];
        kcnt[b]   = (int)sigsum;   // astype(int32): truncate toward zero
        expected_out[b] = sigsum;
    }
}

// ---------------------------------------------------------------------------
// Kernel 2b: rank-based hard top-k + straight-through mask.
// grid (32, B) x 256 threads; each thread ranks one position against the row.
// ---------------------------------------------------------------------------
__global__ __launch_bounds__(256) void k_mask(
    const float* __restrict__ z, const float* __restrict__ rowmax,
    const float* __restrict__ rowsum, const int* __restrict__ kcnt,
    float* __restrict__ mask)
{
    __shared__ float zs[SS];  // 32 KB
    const int b = blockIdx.y;
    const int tid = threadIdx.x;
    const float* zr = z + (size_t)b * SS;
    for (int s = tid; s < SS; s += 256) zs[s] = zr[s];
    __syncthreads();

    const int s0 = blockIdx.x * 256 + tid;
    const float z0 = zs[s0];
    int rank = 0;
    #pragma unroll 4
    for (int s = 0; s < SS; ++s) {
        const float zv = zs[s];
        // stable descending sort: earlier equal elements rank first
        rank += (zv > z0) || (zv == z0 && s < s0);
    }
    const float hard = (rank < kcnt[b]) ? 1.0f : 0.0f;
    const float soft = expf(z0 - rowmax[b]) / rowsum[b];
    mask[(size_t)b * SS + s0] = (hard - soft) + soft;  // match JAX ST arithmetic
}

// ---------------------------------------------------------------------------
// Kernel 3: filtered = TE * mask (bandwidth-dominant pass, float4)
// ---------------------------------------------------------------------------
__global__ __launch_bounds__(256) void k_filter(
    const float* __restrict__ te, const float* __restrict__ mask,
    float* __restrict__ out)
{
    const size_t total4 = (size_t)BB * SS * (EE / 4);
    for (size_t i = (size_t)blockIdx.x * blockDim.x + threadIdx.x; i < total4;
         i += (size_t)gridDim.x * blockDim.x) {
        const float4 v = ((const float4*)te)[i];
        const float mv = mask[i >> 8];  // 256 float4 per token
        float4 o;
        o.x = v.x * mv; o.y = v.y * mv; o.z = v.z * mv; o.w = v.w * mv;
        ((float4*)out)[i] = o;
    }
}

// ---------------------------------------------------------------------------
extern "C" void kernel_launch(void* const* d_in, const int* in_sizes, int n_in,
                              void* d_out, int out_size, void* d_ws, size_t ws_size,
                              hipStream_t stream) {
    const float* te    = (const float*)d_in[0];
    const float* w1    = (const float*)d_in[1];
    const float* bias1 = (const float*)d_in[2];
    const float* w2    = (const float*)d_in[3];
    const float* bias2 = (const float*)d_in[4];
    const float* u     = (const float*)d_in[5];

    float* out         = (float*)d_out;
    float* filtered    = out;                              // B*S*E
    float* maskOut     = out + (size_t)BB * SS * EE;       // B*S
    float* expectedOut = maskOut + (size_t)BB * SS;        // B

    float* ws     = (float*)d_ws;
    float* logits = ws;                       // B*S
    float* z      = ws + (size_t)BB * SS;     // B*S
    float* rowmax = z + (size_t)BB * SS;      // B
    float* rowsum = rowmax + BB;              // B
    int*   kcnt   = (int*)(rowsum + BB);      // B

    const size_t ldsBytes = (size_t)HH * LDS_STRIDE * sizeof(float);  // 263168

    k_logits<<<512, 256, ldsBytes, stream>>>(te, w1, bias1, w2, bias2, logits);
    k_row_stats<<<BB, 1024, 0, stream>>>(logits, u, z, rowmax, rowsum, kcnt, expectedOut);
    k_mask<<<dim3(32, BB), 256, 0, stream>>>(z, rowmax, rowsum, kcnt, maskOut);
    k_filter<<<8192, 256, 0, stream>>>(te, maskOut, filtered);
}